// NodeClassifier_6047313953622
// MI455X (gfx1250) — compile-verified
//
#include <hip/hip_runtime.h>

typedef __attribute__((ext_vector_type(2))) float v2f;
typedef __attribute__((ext_vector_type(8))) float v8f;

#define N_NODES   100000
#define N_EDGES   1600000
#define IN_DIM    128
#define HID_DIM   64
#define N_CLASSES 40
#define M_TILES   (N_NODES / 16)   // 6250 exact 16-row tiles

// LDS B layout: row-pair rp = k/2 holds cols interleaved by k parity:
//   Bs[rp*BSTRIDE + col*2 + (k&1)]
// BSTRIDE = 160 dwords (128 data + 32 pad) => half-wave bank sets are
// disjoint (shift 32 of 64 banks) => conflict-free ds_load_b64 fragments.
#define BSTRIDE 160

// ---------------------------------------------------------------------------
// Zero-fill (graph-capture-safe replacement for hipMemsetAsync)
// ---------------------------------------------------------------------------
__global__ __launch_bounds__(256) void zero_fill_f4(float4* __restrict__ p, int n4) {
    int i = blockIdx.x * 256 + threadIdx.x;
    if (i < n4) p[i] = make_float4(0.f, 0.f, 0.f, 0.f);
}

// ---------------------------------------------------------------------------
// Dense GEMM via V_WMMA_F32_16X16X4_F32:
//   C[M,64] = act(A[M,K]) @ B[K,64] + bias      (act = relu if RELU)
// One wave computes one 16-row M-tile across all four 16-col N-tiles.
// Fragment layouts per CDNA5 ISA 7.12.2:
//   A 16x4 f32 : lane<16 -> VGPR{0,1}=K{0,1}; lane>=16 -> K{2,3}  (b64 global)
//   B 4x16 f32 : VGPR r, lane<16 -> K=r, N=lane; lane>=16 -> K=r+2 (b64 LDS)
//   C 16x16 f32: VGPR r, lane<16 -> M=r, N=lane; lane>=16 -> M=r+8
// ---------------------------------------------------------------------------
template<int K, bool RELU>
__global__ __launch_bounds__(256) void gemm_wmma(const float* __restrict__ A,
                                                 const float* __restrict__ B,
                                                 const float* __restrict__ bias,
                                                 float* __restrict__ C) {
    __shared__ float Bs[(K / 2) * BSTRIDE];
    const int tid = threadIdx.x;
    // Cooperative swizzled stage of B[K][64] into LDS (consecutive-k pairs
    // contiguous per column). Write pattern is bank-conflict-free per wave.
    for (int i = tid; i < K * 64; i += 256) {
        const int k   = i >> 6;
        const int col = i & 63;
        Bs[(k >> 1) * BSTRIDE + col * 2 + (k & 1)] = B[i];
    }
    __syncthreads();

    const int wave = tid >> 5;
    const int lane = tid & 31;
    const int half = lane >> 4;    // which 16-lane half of the wave
    const int l16  = lane & 15;
    const int mt   = blockIdx.x * 8 + wave;

    if (mt < M_TILES) {            // wave-uniform guard: EXEC stays all-1s for WMMA
        const int m0 = mt * 16;
        v8f acc[4] = {};           // 4 N-tiles x 8 VGPRs of f32 accumulators

        const float* arow = A + (size_t)(m0 + l16) * K;
        for (int k0 = 0; k0 < K; k0 += 4) {
            // A fragment: lane holds A[m0+l16][k0+2*half .. +1] (8B-aligned pair)
            float2 av = *(const float2*)(arow + k0 + 2 * half);
            if (RELU) { av.x = fmaxf(av.x, 0.f); av.y = fmaxf(av.y, 0.f); }
            v2f a; a.x = av.x; a.y = av.y;

            const float* brow = &Bs[(k0 / 2 + half) * BSTRIDE + l16 * 2];
#pragma unroll
            for (int nt = 0; nt < 4; ++nt) {
                // Single aligned ds_load_b64 -> even VGPR pair, no repack movs
                const v2f b = *(const v2f*)(brow + nt * 32);
                acc[nt] = __builtin_amdgcn_wmma_f32_16x16x4_f32(
                    /*neg_a=*/false, a, /*neg_b=*/false, b,
                    /*c_mod=*/(short)0, acc[nt],
                    /*reuse_a=*/false, /*reuse_b=*/false);
            }
        }

        // Epilogue: bias add + store (C/D layout: row = m0 + r + 8*half)
#pragma unroll
        for (int nt = 0; nt < 4; ++nt) {
            const int col = nt * 16 + l16;
            const float bv = bias[col];
#pragma unroll
            for (int r = 0; r < 8; ++r) {
                C[(size_t)(m0 + r + 8 * half) * 64 + col] = acc[nt][r] + bv;
            }
        }
    }
}

// ---------------------------------------------------------------------------
// SpMM scatter: out[row[e],:] += val[e] * dense[col[e],:]   (64-wide rows)
// 16 lanes per edge; each lane gathers one float4 (16 lanes = full 256B row,
// coalesced) and issues 4 global_atomic_add_f32 scatters. L2-resident.
// ---------------------------------------------------------------------------
__global__ __launch_bounds__(256) void spmm_scatter(const int*   __restrict__ row,
                                                    const int*   __restrict__ col,
                                                    const float* __restrict__ val,
                                                    const float* __restrict__ dense,
                                                    float*       __restrict__ out) {
    const long long gid = (long long)blockIdx.x * 256 + threadIdx.x;
    const int e = (int)(gid >> 4);
    if (e >= N_EDGES) return;
    const int l = (int)(gid & 15);

    const float v = val[e];
    const int   c = col[e];
    const int   r = row[e];

    const float4 m = ((const float4*)(dense + (size_t)c * HID_DIM))[l];
    float* dst = out + (size_t)r * HID_DIM + l * 4;
    atomicAdd(dst + 0, v * m.x);
    atomicAdd(dst + 1, v * m.y);
    atomicAdd(dst + 2, v * m.z);
    atomicAdd(dst + 3, v * m.w);
}

// ---------------------------------------------------------------------------
// Fused classifier + log-softmax: one thread per node.
// logits = H[node,:] @ Wc + bc ; out = logits - logsumexp(logits)
// Wc/bc broadcast from LDS (uniform indices -> conflict-free broadcast reads).
// ---------------------------------------------------------------------------
__global__ __launch_bounds__(128) void classifier_logsoftmax(const float* __restrict__ H,
                                                             const float* __restrict__ Wc,
                                                             const float* __restrict__ bc,
                                                             float* __restrict__ out) {
    __shared__ float Ws[HID_DIM * N_CLASSES];
    __shared__ float bs[N_CLASSES];
    const int tid = threadIdx.x;
    for (int i = tid; i < HID_DIM * N_CLASSES; i += 128) Ws[i] = Wc[i];
    if (tid < N_CLASSES) bs[tid] = bc[tid];
    __syncthreads();

    const int node = blockIdx.x * 128 + tid;
    if (node >= N_NODES) return;

    float acc[N_CLASSES];
#pragma unroll
    for (int c = 0; c < N_CLASSES; ++c) acc[c] = bs[c];

    const float* h = H + (size_t)node * HID_DIM;
    for (int k = 0; k < HID_DIM; ++k) {
        const float hv = h[k];
#pragma unroll
        for (int c = 0; c < N_CLASSES; ++c)
            acc[c] = fmaf(hv, Ws[k * N_CLASSES + c], acc[c]);
    }

    float mx = acc[0];
#pragma unroll
    for (int c = 1; c < N_CLASSES; ++c) mx = fmaxf(mx, acc[c]);
    float s = 0.f;
#pragma unroll
    for (int c = 0; c < N_CLASSES; ++c) s += expf(acc[c] - mx);
    const float lse = mx + logf(s);

    float* o = out + (size_t)node * N_CLASSES;
#pragma unroll
    for (int c = 0; c < N_CLASSES; ++c) o[c] = acc[c] - lse;
}

// ---------------------------------------------------------------------------
// Launch: x@W1+b1 -> spmm -> relu@W2+b2 -> spmm -> classifier+logsoftmax
// Workspace ping-pong: buf0 = support/dense, buf1 = aggregation target.
// ---------------------------------------------------------------------------
extern "C" void kernel_launch(void* const* d_in, const int* in_sizes, int n_in,
                              void* d_out, int out_size, void* d_ws, size_t ws_size,
                              hipStream_t stream) {
    const float* x    = (const float*)d_in[0];
    const int*   erow = (const int*)  d_in[1];
    const int*   ecol = (const int*)  d_in[2];
    const float* eval = (const float*)d_in[3];
    const float* W1   = (const float*)d_in[4];
    const float* b1   = (const float*)d_in[5];
    const float* W2   = (const float*)d_in[6];
    const float* b2   = (const float*)d_in[7];
    const float* Wc   = (const float*)d_in[8];
    const float* bc   = (const float*)d_in[9];
    float* out = (float*)d_out;

    float* buf0 = (float*)d_ws;                        // [N_NODES * 64]
    float* buf1 = buf0 + (size_t)N_NODES * HID_DIM;    // [N_NODES * 64]

    const int gemm_grid = (M_TILES + 7) / 8;                      // 782 blocks, 8 waves each
    const int spmm_grid = (int)(((long long)N_EDGES * 16) / 256); // 100000 blocks
    const int n4        = N_NODES * HID_DIM / 4;
    const int zero_grid = (n4 + 255) / 256;

    // Layer 1: support = x @ W1 + b1
    gemm_wmma<IN_DIM, false><<<gemm_grid, 256, 0, stream>>>(x, W1, b1, buf0);
    // h1_agg = A @ support
    zero_fill_f4<<<zero_grid, 256, 0, stream>>>((float4*)buf1, n4);
    spmm_scatter<<<spmm_grid, 256, 0, stream>>>(erow, ecol, eval, buf0, buf1);
    // Layer 2: support = relu(h1_agg) @ W2 + b2   (relu fused into A-load)
    gemm_wmma<HID_DIM, true><<<gemm_grid, 256, 0, stream>>>(buf1, W2, b2, buf0);
    // h2_agg = A @ support
    zero_fill_f4<<<zero_grid, 256, 0, stream>>>((float4*)buf1, n4);
    spmm_scatter<<<spmm_grid, 256, 0, stream>>>(erow, ecol, eval, buf0, buf1);
    // Classifier + log-softmax
    classifier_logsoftmax<<<(N_NODES + 127) / 128, 128, 0, stream>>>(buf1, Wc, bc, out);
}